// RQV_33500744909203
// MI455X (gfx1250) — compile-verified
//
#include <hip/hip_runtime.h>
#include <hip/hip_bf16.h>

// ---------------------------------------------------------------------------
// Residual Vector Quantizer (8 stages, EMA codebook update) for MI455X gfx1250
// Distance GEMM on the fp32 WMMA path: v_wmma_f32_16x16x4_f32, with B-fragment
// register preload (back-to-back WMMA issue) and global->LDS staging of the
// next code chunk overlapped with the matrix ops.
// ---------------------------------------------------------------------------

#define NQ     8
#define NC     1024
#define DD     128
#define BB     16
#define SS     2048
#define ROWS   (BB * SS)          // 32768 vectors
#define GAMMA_ 0.99f
#define ONEMG_ 0.01f

typedef float v2f __attribute__((ext_vector_type(2)));
typedef float v8f __attribute__((ext_vector_type(8)));

// ------------------------------- utility -----------------------------------

__global__ void rqv_zero_f32(float* __restrict__ p, size_t n) {
  size_t i = (size_t)blockIdx.x * blockDim.x + threadIdx.x;
  size_t stride = (size_t)gridDim.x * blockDim.x;
  for (; i < n; i += stride) p[i] = 0.0f;
}

// data [b, D, s]  ->  R [b*s, D]   (tiled, coalesced both sides)
__global__ __launch_bounds__(256)
void rqv_transpose_in(const float* __restrict__ data, float* __restrict__ R) {
  __shared__ float tile[32][33];
  const int b  = blockIdx.z;
  const int s0 = blockIdx.x * 32;
  const int d0 = blockIdx.y * 32;
  const int tx = threadIdx.x, ty = threadIdx.y;      // 32 x 8
#pragma unroll
  for (int j = 0; j < 4; ++j) {
    int d = d0 + ty + j * 8;
    tile[ty + j * 8][tx] = data[((size_t)b * DD + d) * SS + s0 + tx];
  }
  __syncthreads();
#pragma unroll
  for (int j = 0; j < 4; ++j) {
    int s = s0 + ty + j * 8;
    R[((size_t)b * SS + s) * DD + d0 + tx] = tile[tx][ty + j * 8];
  }
}

// cur_y [b*s, D] -> logits [b, D, s], fused commitment-loss accumulation
__global__ __launch_bounds__(256)
void rqv_transpose_out_loss(const float* __restrict__ cury,
                            const float* __restrict__ data,
                            float* __restrict__ logits,
                            float* __restrict__ loss_acc) {
  __shared__ float tile[32][33];
  __shared__ float part[8];
  const int b  = blockIdx.z;
  const int s0 = blockIdx.x * 32;
  const int d0 = blockIdx.y * 32;
  const int tx = threadIdx.x, ty = threadIdx.y;
#pragma unroll
  for (int j = 0; j < 4; ++j) {
    int s = s0 + ty + j * 8;
    tile[ty + j * 8][tx] = cury[((size_t)b * SS + s) * DD + d0 + tx];
  }
  __syncthreads();
  float sq = 0.0f;
#pragma unroll
  for (int j = 0; j < 4; ++j) {
    int d = d0 + ty + j * 8;
    size_t o = ((size_t)b * DD + d) * SS + s0 + tx;
    float v = tile[tx][ty + j * 8];
    logits[o] = v;                       // logits == cur_y_t numerically
    float df = v - data[o];
    sq += df * df;
  }
  const int t = ty * 32 + tx;
  const int lane = t & 31, wave = t >> 5;
#pragma unroll
  for (int m = 16; m >= 1; m >>= 1) sq += __shfl_down(sq, m, 32);
  if (lane == 0) part[wave] = sq;
  __syncthreads();
  if (t == 0) {
    float s = 0.0f;
#pragma unroll
    for (int w = 0; w < 8; ++w) s += part[w];
    atomicAdd(loss_acc, s);
  }
}

__global__ void rqv_finalize_loss(const float* __restrict__ loss_acc,
                                  float* __restrict__ out) {
  out[0] = loss_acc[0] * (1.0f / (float)((size_t)BB * DD * SS));
}

__global__ void rqv_idx_to_float(const int* __restrict__ idx,
                                 float* __restrict__ out, int n) {
  int i = blockIdx.x * blockDim.x + threadIdx.x;
  int stride = gridDim.x * blockDim.x;
  for (; i < n; i += stride) out[i] = (float)idx[i];
}

// ---------------------------- per-stage kernels -----------------------------

// half ||w||^2 per code; zero counts + sums.  grid=NC, block=128
__global__ __launch_bounds__(128)
void rqv_stage_prep(const float* __restrict__ W, float* __restrict__ halfw2,
                    float* __restrict__ counts, float* __restrict__ sums) {
  __shared__ float part[4];
  const int n = blockIdx.x, d = threadIdx.x;
  const int lane = d & 31, wave = d >> 5;
  float w = W[(size_t)n * DD + d];
  float sq = w * w;
#pragma unroll
  for (int m = 16; m >= 1; m >>= 1) sq += __shfl_down(sq, m, 32);
  if (lane == 0) part[wave] = sq;
  sums[(size_t)n * DD + d] = 0.0f;
  __syncthreads();
  if (d == 0) {
    halfw2[n] = 0.5f * (part[0] + part[1] + part[2] + part[3]);
    counts[n] = 0.0f;
  }
}

// ---------------------------------------------------------------------------
// Nearest-code search via fp32 WMMA.
//   score(row, n) = dot(R[row], W[n]) - 0.5*||W[n]||^2 ;  idx = argmax score
// Block = 128 threads (4 waves), owns 64 rows; sweeps 1024 codes in 16-code
// chunks. A tile register-resident; B chunk fragments preloaded into registers
// so the 32 WMMAs issue back-to-back, while the NEXT chunk is staged
// global->LDS concurrently (single-buffer software pipeline).
// ---------------------------------------------------------------------------
#define LDSA_PITCH 130   // 130 % 64 == 2 -> conflict-free 16-row column reads
#define LDSB_PITCH 130
#define NCHUNK     (NC / 16)

__global__ __launch_bounds__(128)
void rqv_argmin_wmma(const float* __restrict__ R, const float* __restrict__ W,
                     const float* __restrict__ halfw2, int* __restrict__ idxout) {
  __shared__ float ldsA[64 * LDSA_PITCH];
  __shared__ float ldsB[16 * LDSB_PITCH];

  const int t     = threadIdx.x;
  const int lane  = t & 31;
  const int wave  = t >> 5;
  const int hi    = lane >> 4;           // 0: K even pair, 1: K odd pair
  const int lrow  = lane & 15;
  const int rbase = blockIdx.x * 64;

  // cooperative staging thread-slice (used only by the W prefetch)
  const int srow4 = t >> 2;              // 0..31  (prefetch: W row = n0 + 32 + srow4)
  const int sc4   = (t & 3) << 2;        // float4 column base within 16-col half

  // ---- cooperative, coalesced load of 64x128 A tile into LDS --------------
  for (int it = t; it < 64 * 32; it += 128) {
    int row = it >> 5;
    int c4  = (it & 31) << 2;
    const float4 v = *(const float4*)&R[(size_t)(rbase + row) * DD + c4];
    float* dst = &ldsA[row * LDSA_PITCH + c4];
    dst[0] = v.x; dst[1] = v.y; dst[2] = v.z; dst[3] = v.w;
  }
  __syncthreads();

  // ---- pull this lane's A fragments into registers ------------------------
  // 16x4 f32 A layout: lanes 0-15 hold K = kk,kk+1 ; lanes 16-31 K = kk+2,kk+3
  v2f a[32];
  {
    const int m = wave * 16 + lrow;
#pragma unroll
    for (int kk = 0; kk < 32; ++kk) {
      int c = kk * 4 + hi * 2;
      a[kk].x = ldsA[m * LDSA_PITCH + c];
      a[kk].y = ldsA[m * LDSA_PITCH + c + 1];
    }
  }

  float best[8];
  int   bidx[8];
#pragma unroll
  for (int r = 0; r < 8; ++r) { best[r] = -3.4e38f; bidx[r] = 0; }

  // ---- prologue: stage code chunk 0 into ldsB ------------------------------
  {
    for (int it = t; it < 16 * 32; it += 128) {
      int row = it >> 5;
      int c4  = (it & 31) << 2;
      const float4 v = *(const float4*)&W[(size_t)row * DD + c4];
      float* dst = &ldsB[row * LDSB_PITCH + c4];
      dst[0] = v.x; dst[1] = v.y; dst[2] = v.z; dst[3] = v.w;
    }
  }

  // ---- sweep the 1024 codes in 16-wide chunks ------------------------------
  for (int nc = 0; nc < NCHUNK; ++nc) {
    const int n0 = nc * 16;
    __syncthreads();                       // chunk nc resident in ldsB

    // preload ALL 32 B fragments for this chunk into registers
    v2f bf[32];
#pragma unroll
    for (int kk = 0; kk < 32; ++kk) {
      int c = kk * 4 + hi * 2;
      bf[kk].x = ldsB[lrow * LDSB_PITCH + c];
      bf[kk].y = ldsB[lrow * LDSB_PITCH + c + 1];
    }
    const int   col = n0 + lrow;
    const float hw  = halfw2[col];
    __syncthreads();                       // all waves done reading ldsB

    // stage chunk nc+1 into ldsB -- overlaps with the WMMA burst below
    if (nc + 1 < NCHUNK) {
      for (int it = t; it < 16 * 32; it += 128) {
        int row = it >> 5;
        int c4  = (it & 31) << 2;
        const float4 v =
            *(const float4*)&W[(size_t)(n0 + 16 + row) * DD + c4];
        float* dst = &ldsB[row * LDSB_PITCH + c4];
        dst[0] = v.x; dst[1] = v.y; dst[2] = v.z; dst[3] = v.w;
      }
      if (nc + 2 < NCHUNK) {               // warm L2/WGP$ for chunk nc+2
        __builtin_prefetch(&W[(size_t)(n0 + 32 + srow4) * DD + sc4], 0, 1);
      }
    }

    // 32 back-to-back fp32 WMMAs (K = 128 in steps of 4)
    v8f acc = {0.f, 0.f, 0.f, 0.f, 0.f, 0.f, 0.f, 0.f};
#pragma unroll
    for (int kk = 0; kk < 32; ++kk) {
      acc = __builtin_amdgcn_wmma_f32_16x16x4_f32(
          /*neg_a=*/false, a[kk], /*neg_b=*/false, bf[kk],
          /*c_mod=*/(short)0, acc, /*reuse_a=*/false, /*reuse_b=*/false);
    }

#pragma unroll
    for (int r = 0; r < 8; ++r) {
      float sc = acc[r] - hw;
      if (sc > best[r]) { best[r] = sc; bidx[r] = col; }   // '>' keeps first
    }
  }

  // ---- cross-lane argmax over the 16 lanes that share a row ----------------
#pragma unroll
  for (int m = 1; m < 16; m <<= 1) {
#pragma unroll
    for (int r = 0; r < 8; ++r) {
      float ov = __shfl_xor(best[r], m, 32);
      int   oi = __shfl_xor(bidx[r], m, 32);
      if (ov > best[r] || (ov == best[r] && oi < bidx[r])) {
        best[r] = ov; bidx[r] = oi;
      }
    }
  }
  if (lrow == 0) {        // lane 0 -> rows 0..7 ; lane 16 -> rows 8..15
    int rowb = rbase + wave * 16 + hi * 8;
#pragma unroll
    for (int r = 0; r < 8; ++r) idxout[rowb + r] = bidx[r];
  }
}

// cur_y (=|+=) Q ; R_next = R - Q ; scatter counts/sums.  grid=ROWS, block=128
__global__ __launch_bounds__(128)
void rqv_update(const float* __restrict__ Rold, float* __restrict__ Rnew,
                float* __restrict__ cury, const float* __restrict__ W,
                const int* __restrict__ idx, float* __restrict__ counts,
                float* __restrict__ sums, int first) {
  const int row = blockIdx.x, d = threadIdx.x;
  const int code = idx[row];
  const float q  = W[(size_t)code * DD + d];
  const float ro = Rold[(size_t)row * DD + d];
  cury[(size_t)row * DD + d] = first ? q : (cury[(size_t)row * DD + d] + q);
  Rnew[(size_t)row * DD + d] = ro - q;
  atomicAdd(&sums[(size_t)code * DD + d], ro);
  if (d == 0) atomicAdd(&counts[code], 1.0f);
}

// EMA codebook update + dead-code replacement.  grid=NC, block=128
__global__ __launch_bounds__(128)
void rqv_ema(const float* __restrict__ Ni, const float* __restrict__ mi,
             const float* __restrict__ counts, const float* __restrict__ sums,
             const float* __restrict__ Rold, int stage,
             float* __restrict__ outNi, float* __restrict__ outMi,
             float* __restrict__ outW) {
  const int n = blockIdx.x, d = threadIdx.x;
  const float cnt    = counts[n];
  const float ni_new = Ni[n] * GAMMA_ + cnt * ONEMG_;
  const float mi_new = mi[(size_t)n * DD + d] * GAMMA_ +
                       sums[(size_t)n * DD + d] * ONEMG_;
  float w = mi_new / fmaxf(ni_new, 1e-8f);
  if (ni_new < 2.0f) {
    unsigned rid = ((unsigned)n * 2654435761u + (unsigned)stage * 40503u
                    + 12345u) % (unsigned)ROWS;
    w = Rold[(size_t)rid * DD + d];
  }
  if (d == 0) outNi[n] = ni_new;
  outMi[(size_t)n * DD + d] = mi_new;
  outW[(size_t)n * DD + d]  = w;
}

// ------------------------------- host glue ----------------------------------

extern "C" void kernel_launch(void* const* d_in, const int* in_sizes, int n_in,
                              void* d_out, int out_size, void* d_ws, size_t ws_size,
                              hipStream_t stream) {
  const float* data    = (const float*)d_in[0];   // [16, 128, 2048]
  const float* weights = (const float*)d_in[1];   // [8, 1024, 128]
  const float* Ni      = (const float*)d_in[2];   // [8, 1024]
  const float* mi      = (const float*)d_in[3];   // [8, 1024, 128]
  float* out = (float*)d_out;

  // output layout (flat floats, return order)
  const size_t LOGITS_OFF = 0;
  const size_t LOSS_OFF   = (size_t)BB * DD * SS;            // 4194304
  const size_t IDX_OFF    = LOSS_OFF + 1;
  const size_t NI_OFF     = IDX_OFF + (size_t)NQ * ROWS;     // +262144
  const size_t MI_OFF     = NI_OFF + (size_t)NQ * NC;        // +8192
  const size_t W_OFF      = MI_OFF + (size_t)NQ * NC * DD;   // +1048576

  // workspace carve-up
  char* w = (char*)d_ws;
  float* R0     = (float*)w; w += (size_t)ROWS * DD * sizeof(float);
  float* R1     = (float*)w; w += (size_t)ROWS * DD * sizeof(float);
  float* cury   = (float*)w; w += (size_t)ROWS * DD * sizeof(float);
  float* sums   = (float*)w; w += (size_t)NC * DD * sizeof(float);
  float* counts = (float*)w; w += (size_t)NC * sizeof(float);
  float* halfw2 = (float*)w; w += (size_t)NC * sizeof(float);
  int*   idxbuf = (int*)w;   w += (size_t)NQ * ROWS * sizeof(int);
  float* lossa  = (float*)w; w += sizeof(float);

  // init: loss = 0, residual = transpose(data)
  rqv_zero_f32<<<1, 32, 0, stream>>>(lossa, 1);
  dim3 tgrid(SS / 32, DD / 32, BB), tblk(32, 8);
  rqv_transpose_in<<<tgrid, tblk, 0, stream>>>(data, R0);

  float* Rcur = R0;
  float* Rnxt = R1;
  for (int i = 0; i < NQ; ++i) {
    const float* Wst = weights + (size_t)i * NC * DD;
    rqv_stage_prep<<<NC, 128, 0, stream>>>(Wst, halfw2, counts, sums);
    rqv_argmin_wmma<<<ROWS / 64, 128, 0, stream>>>(Rcur, Wst, halfw2,
                                                   idxbuf + (size_t)i * ROWS);
    rqv_update<<<ROWS, 128, 0, stream>>>(Rcur, Rnxt, cury, Wst,
                                         idxbuf + (size_t)i * ROWS,
                                         counts, sums, (i == 0) ? 1 : 0);
    rqv_ema<<<NC, 128, 0, stream>>>(Ni + (size_t)i * NC,
                                    mi + (size_t)i * NC * DD,
                                    counts, sums, Rcur, i,
                                    out + NI_OFF + (size_t)i * NC,
                                    out + MI_OFF + (size_t)i * NC * DD,
                                    out + W_OFF + (size_t)i * NC * DD);
    float* tmp = Rcur; Rcur = Rnxt; Rnxt = tmp;
  }

  rqv_transpose_out_loss<<<tgrid, tblk, 0, stream>>>(cury, data,
                                                     out + LOGITS_OFF, lossa);
  rqv_finalize_loss<<<1, 1, 0, stream>>>(lossa, out + LOSS_OFF);
  rqv_idx_to_float<<<512, 256, 0, stream>>>(idxbuf, out + IDX_OFF,
                                            NQ * ROWS);
}